// InnerCausalLinearAttention_88450556493922
// MI455X (gfx1250) — compile-verified
//
#include <hip/hip_runtime.h>

// ---------------------------------------------------------------------------
// Causal linear attention (linear transformer), chunked parallel-scan form.
//   phi(x) = elu(x)+1 ;  S_i = sum_{j<=i} phi(k_j) v_j^T ;
//   out_i = phi(q_i) S_i / (phi(q_i) . sum_{j<=i} phi(k_j) + eps)
// Shapes: N=2, L=2048, H=8, D=64, M=64, fp32.
// Three kernels:
//   k1: per-chunk local state  G_c = phi(K_c)^T @ [V_c | 1]   (WMMA f32)
//   k2: exclusive prefix over chunks (elementwise, massively parallel)
//   k3: O = mask(phi(Q)phi(K)^T) @ [V|1] + phi(Q) @ P_c, normalize (WMMA f32)
// The appended ones-column makes the normalizer z fall out of the same GEMMs.
// fp32 WMMA (16x16x4) is used on purpose: bf16 operands would accrue
// ~sqrt(L)*2^-8 ~ 18% relative error over 2048-term causal sums.
// ---------------------------------------------------------------------------

typedef __attribute__((ext_vector_type(2))) float v2f;
typedef __attribute__((ext_vector_type(8))) float v8f;

#define EPSV 1e-6f
#define NB   2
#define LSEQ 2048
#define NHD  8
#define DD   64
#define NSEQ (NB * NHD)        // 16 head-sequences
#define CHUNK 64
#define NC   (LSEQ / CHUNK)    // 32 chunks per sequence
#define MP   80                // padded state cols: 64 value cols + ones col + pad
#define ROWSTRIDE (NHD * DD)   // 512 floats between consecutive l, fixed (n,h)

// Required workspace: NSEQ*NC*DD*MP*4 = 10,485,760 bytes (chunk states).

__device__ __forceinline__ float phi(float x) {
    // elu(x) + 1 : strictly positive feature map
    return x > 0.0f ? x + 1.0f : __expf(x);
}

// 8-byte vector load of a contiguous column pair (global or LDS).
struct alignas(8) F2 { float x, y; };
__device__ __forceinline__ F2 ld2(const float* p) {
    return *reinterpret_cast<const F2*>(p);
}

// D = A(16x4 f32) * B(4x16 f32) + C(16x16 f32), one wave32.
__device__ __forceinline__ v8f wmma_f32(v2f a, v2f b, v8f c) {
    return __builtin_amdgcn_wmma_f32_16x16x4_f32(
        /*neg_a=*/false, a, /*neg_b=*/false, b,
        /*c_mod=*/(short)0, c, /*reuse_a=*/false, /*reuse_b=*/false);
}

// ---------------------------------------------------------------------------
// Kernel 1: per-chunk local KV state.  grid = NSEQ*NC blocks, 128 threads.
// Wave w owns d-tile w (rows 16w..16w+15 of the 64x80 state).
// ---------------------------------------------------------------------------
__global__ void __launch_bounds__(128)
k1_chunk_kv(const float* __restrict__ Kp, const float* __restrict__ Vp,
            float* __restrict__ G) {
    const int c  = blockIdx.x & (NC - 1);
    const int s  = blockIdx.x >> 5;          // (n,h) pair
    const int n  = s >> 3, h = s & 7;
    const int lane = threadIdx.x & 31;
    const int w    = threadIdx.x >> 5;       // d-tile index 0..3
    const int lo = lane & 15, hi = lane >> 4;
    const size_t base = ((size_t)n * LSEQ * NHD + h) * DD;
    const int l0 = c * CHUNK;

    // A-operand fragments of phi(K)^T, tile rows d = 16w..16w+15:
    // ka[jj].r = phi(K[l0 + 4*jj + r + 2*hi][16*w + lo])   (row-strided: b32)
    v2f ka[16];
#pragma unroll
    for (int jj = 0; jj < 16; ++jj) {
        size_t r0 = (size_t)(l0 + 4 * jj + 2 * hi);
        ka[jj].x = phi(Kp[base + r0 * ROWSTRIDE + 16 * w + lo]);
        ka[jj].y = phi(Kp[base + (r0 + 1) * ROWSTRIDE + 16 * w + lo]);
    }

    const size_t gbase = (size_t)(s * NC + c) * DD * MP;
#pragma unroll
    for (int mt = 0; mt < 5; ++mt) {         // 4 value m-tiles + ones-column tile
        v8f acc = {};
#pragma unroll
        for (int jj = 0; jj < 16; ++jj) {
            v2f b;
            if (mt < 4) {
                size_t r0 = (size_t)(l0 + 4 * jj + 2 * hi);
                b.x = Vp[base + r0 * ROWSTRIDE + 16 * mt + lo];
                b.y = Vp[base + (r0 + 1) * ROWSTRIDE + 16 * mt + lo];
            } else {                          // V' column 64 == 1, rest 0
                b.x = (lo == 0) ? 1.0f : 0.0f;
                b.y = b.x;
            }
            acc = wmma_f32(ka[jj], b, acc);
        }
#pragma unroll
        for (int r = 0; r < 8; ++r) {
            int d = 16 * w + r + 8 * hi;
            G[gbase + (size_t)d * MP + 16 * mt + lo] = acc[r];
        }
    }
}

// ---------------------------------------------------------------------------
// Kernel 2: in-place exclusive prefix sum over the NC chunk states.
// One thread per (sequence, state element); consecutive tids -> consecutive m
// (coalesced).
// ---------------------------------------------------------------------------
__global__ void k2_scan(float* __restrict__ G) {
    const int per_seq = DD * MP;                       // 5120
    int tid = blockIdx.x * blockDim.x + threadIdx.x;
    if (tid >= NSEQ * per_seq) return;
    int s = tid / per_seq;
    int e = tid - s * per_seq;
    size_t idx = (size_t)s * NC * per_seq + e;
    float run = 0.0f;
    for (int c = 0; c < NC; ++c) {
        float g = G[idx];
        G[idx]  = run;                                 // exclusive prefix
        run    += g;
        idx    += per_seq;
    }
}

// ---------------------------------------------------------------------------
// Kernel 3: per-chunk output.  grid = NSEQ*NC blocks, 128 threads.
// Wave w owns output row-tile w (rows l0+16w .. l0+16w+15).
// ---------------------------------------------------------------------------
__global__ void __launch_bounds__(128)
k3_out(const float* __restrict__ Qp, const float* __restrict__ Kp,
       const float* __restrict__ Vp, const float* __restrict__ P,
       float* __restrict__ Out) {
    __shared__ float As[4][4][16 * 16];      // [wave][j-tile][row*16+col]
    const int c  = blockIdx.x & (NC - 1);
    const int s  = blockIdx.x >> 5;
    const int n  = s >> 3, h = s & 7;
    const int lane = threadIdx.x & 31;
    const int w    = threadIdx.x >> 5;
    const int lo = lane & 15, hi = lane >> 4;
    const size_t base = ((size_t)n * LSEQ * NHD + h) * DD;
    const int l0 = c * CHUNK;
    const size_t pbase = (size_t)(s * NC + c) * DD * MP;

    // Warm the prefix-state (20 KB, cold HBM) while the A-phase WMMAs run:
    // each of 128 threads prefetches one 160 B slice -> covers all 64*MP*4 B.
    __builtin_prefetch(&P[pbase + (size_t)threadIdx.x * 40], 0, 3);

    // phi(Q) A-fragments for row tile w, all 16 K-steps over d (b64 pairs):
    v2f qa[16];
    {
        const float* qrow = Qp + base + (size_t)(l0 + 16 * w + lo) * ROWSTRIDE;
#pragma unroll
        for (int dd = 0; dd < 16; ++dd) {
            F2 t = ld2(qrow + 4 * dd + 2 * hi);
            qa[dd].x = phi(t.x);
            qa[dd].y = phi(t.y);
        }
    }

    // A = phi(Q)phi(K)^T row tiles (w, jt<=w); causal mask on diagonal tile.
    for (int jt = 0; jt <= w; ++jt) {
        const float* krow = Kp + base + (size_t)(l0 + 16 * jt + lo) * ROWSTRIDE;
        v8f a = {};
#pragma unroll
        for (int dd = 0; dd < 16; ++dd) {
            F2 t = ld2(krow + 4 * dd + 2 * hi);  // B-frag of K^T == row-major K
            v2f b;
            b.x = phi(t.x);
            b.y = phi(t.y);
            a = wmma_f32(qa[dd], b, a);
        }
        const bool diag = (jt == w);
#pragma unroll
        for (int r = 0; r < 8; ++r) {
            int Mrow = r + 8 * hi;
            float v = a[r];
            if (diag && lo > Mrow) v = 0.0f; // keep j <= i
            As[w][jt][Mrow * 16 + lo] = v;
        }
    }
    __syncthreads();                          // LDS visibility for A-frag reads

    v8f o[5];
#pragma unroll
    for (int mt = 0; mt < 5; ++mt) {
        v8f acc = {};
        // inter-chunk: phi(Q) @ P_c  (P holds prefix state incl. ksum column)
#pragma unroll
        for (int dd = 0; dd < 16; ++dd) {
            v2f b;                            // row-strided (320 B): b32 pair
            b.x = P[pbase + (size_t)(4 * dd + 2 * hi) * MP + 16 * mt + lo];
            b.y = P[pbase + (size_t)(4 * dd + 2 * hi + 1) * MP + 16 * mt + lo];
            acc = wmma_f32(qa[dd], b, acc);
        }
        // intra-chunk: mask(A) @ [V | 1]
        for (int jt = 0; jt <= w; ++jt) {
            const float* arow = &As[w][jt][lo * 16];
#pragma unroll
            for (int jj = 0; jj < 4; ++jj) {
                F2 t = ld2(arow + 4 * jj + 2 * hi);   // ds_load_b64
                v2f a; a.x = t.x; a.y = t.y;
                v2f b;
                if (mt < 4) {
                    size_t r0 = (size_t)(l0 + 16 * jt + 4 * jj + 2 * hi);
                    b.x = Vp[base + r0 * ROWSTRIDE + 16 * mt + lo];
                    b.y = Vp[base + (r0 + 1) * ROWSTRIDE + 16 * mt + lo];
                } else {
                    b.x = (lo == 0) ? 1.0f : 0.0f;
                    b.y = b.x;
                }
                acc = wmma_f32(a, b, acc);
            }
        }
        o[mt] = acc;
    }

    // Normalizer z[row] = o[4] column 0 (lanes 0 / 16); broadcast + scale.
#pragma unroll
    for (int r = 0; r < 8; ++r) {
        float z  = __shfl(o[4][r], hi << 4, 32);
        float rz = 1.0f / (z + EPSV);
        int l = l0 + 16 * w + r + 8 * hi;
        size_t rowoff = base + (size_t)l * ROWSTRIDE;
#pragma unroll
        for (int mt = 0; mt < 4; ++mt)
            Out[rowoff + 16 * mt + lo] = o[mt][r] * rz;
    }
}

// ---------------------------------------------------------------------------
extern "C" void kernel_launch(void* const* d_in, const int* in_sizes, int n_in,
                              void* d_out, int out_size, void* d_ws, size_t ws_size,
                              hipStream_t stream) {
    const float* Q = (const float*)d_in[0];
    const float* K = (const float*)d_in[1];
    const float* V = (const float*)d_in[2];
    float* G   = (float*)d_ws;   // NSEQ*NC*DD*MP floats = 10 MB chunk states
    float* Out = (float*)d_out;

    dim3 blk(128);
    k1_chunk_kv<<<NSEQ * NC, blk, 0, stream>>>(K, V, G);

    int scan_threads = NSEQ * DD * MP;   // 81920
    k2_scan<<<(scan_threads + 255) / 256, 256, 0, stream>>>(G);

    k3_out<<<NSEQ * NC, blk, 0, stream>>>(Q, K, V, G, Out);
}